// Model_57784490000647
// MI455X (gfx1250) — compile-verified
//
#include <hip/hip_runtime.h>
#include <math.h>

typedef __attribute__((ext_vector_type(2))) float v2f;
typedef __attribute__((ext_vector_type(8))) float v8f;

#define HH 128
#define WW 128
#define BB 16
#define HW (HH * WW)

// ---------------------------------------------------------------------------
// Implicit-GEMM 3x3 SAME conv as 9 shifted GEMMs, NCHW, f32, using
// V_WMMA_F32_16X16X4_F32.  Outer loop: kernel position pos=(kh,kw); inner
// K dimension is plain ci, so A addresses are base + ci*HW with constant
// unrolled immediate offsets (no divisions, no exec branching: halo handled
// by clamped address * 0/1 mask, constant per pos).
//
// LDS B layout is packed as [k>>2][co][4] (16B per (k-group, co)) so each
// WMMA B operand {k,k+1}/{k+2,k+3} is one aligned ds_load_b64 straight into
// an even VGPR pair: one DS op per WMMA, no register re-pairing movs, and a
// single address register (co*16 + koff*4) with constant immediate offsets.
//
// M = 16-pixel strip along W, N = Cout padded to 16. One wave accumulates
// ALL Cout tiles (<=10 x v8f) so each A fragment is reused across N tiles.
// ACT: 0 = ReLU, 1 = sigmoid.
// ---------------------------------------------------------------------------
template<int CIN, int COUT, int ACT>
__global__ __launch_bounds__(128)
void conv3x3_wmma(const float* __restrict__ x, const float* __restrict__ wr,
                  const float* __restrict__ bias, float* __restrict__ y) {
  constexpr int COUTP = (COUT + 15) & ~15;
  constexpr int NT    = COUTP / 16;
  // ci chunk size: must divide CIN and be a multiple of 4 (full unroll,
  // constant-immediate offsets). 156->52, 128/64->32, 8->8.
  constexpr int KC    = (CIN % 32 == 0) ? 32 : ((CIN % 52 == 0) ? 52 : CIN);
  static_assert(CIN % KC == 0 && KC % 4 == 0, "bad KC");

  __shared__ float sW[KC * COUTP];        // packed: [(k>>2)][co][k&3]
  __shared__ float sO[4 * COUTP * 16];    // per-wave output staging [co][m]

  const int tid  = threadIdx.x;
  const int wave = tid >> 5;
  const int lane = tid & 31;

  const int mt   = blockIdx.x * 4 + wave; // 16-pixel strip id
  const int pix0 = mt * 16;
  const int w0   = pix0 % WW;
  const int h    = (pix0 / WW) % HH;
  const int n    = pix0 / (WW * HH);

  const int m    = lane & 15;             // pixel-in-strip / N-in-tile
  const int koff = (lane < 16) ? 0 : 2;   // ISA A/B f32 lane-half K split
  const int wpix = w0 + m;

  const float* xn = x + (size_t)n * CIN * HW;

  v8f acc[NT];
  #pragma unroll
  for (int t = 0; t < NT; ++t) acc[t] = (v8f)0.f;

  for (int pos = 0; pos < 9; ++pos) {
    const int kh = pos / 3;               // small unrollable loop: constants
    const int kw = pos - kh * 3;
    const int hh = h + kh - 1;
    const int ww = wpix + kw - 1;
    const bool vld = (hh >= 0) & (hh < HH) & (ww >= 0) & (ww < WW);
    const float msk = vld ? 1.f : 0.f;
    const int hc = hh < 0 ? 0 : (hh >= HH ? HH - 1 : hh);
    const int wc = ww < 0 ? 0 : (ww >= WW ? WW - 1 : ww);
    const float* xb = xn + hc * WW + wc;  // ci = 0, clamped in-bounds

    for (int cc = 0; cc < CIN; cc += KC) {
      __syncthreads();
      // weight stage with CONTIGUOUS (conflict-free) LDS writes:
      // sW[pidx], pidx = ((cl>>2)*COUTP + co)*4 + (cl&3)
      for (int pidx = tid; pidx < KC * COUTP; pidx += 128) {
        const int j  = pidx & 3;               // k within group of 4
        const int co = (pidx >> 2) % COUTP;
        const int kg = pidx / (4 * COUTP);
        const int cl = kg * 4 + j;
        float v = 0.f;
        if (co < COUT)
          v = wr[((size_t)pos * CIN + cc + cl) * COUT + co];
        sW[pidx] = v;
      }
      __syncthreads();

      const float* xp = xb + (size_t)(cc + koff) * HW;

      #pragma unroll
      for (int cs = 0; cs < KC; cs += 4) {
        // ---- A fragment: v0 = K{0,2}, v1 = K{1,3} by lane half ----
        v2f a;
        a[0] = xp[(size_t)cs * HW] * msk;        // constant-offset loads
        a[1] = xp[(size_t)(cs + 1) * HW] * msk;
        // ---- per N-tile: one aligned ds_load_b64 B fragment + WMMA ----
        #pragma unroll
        for (int t = 0; t < NT; ++t) {
          const int co = t * 16 + m;
          const v2f b = *(const v2f*)&sW[((cs >> 2) * COUTP + co) * 4 + koff];
          acc[t] = __builtin_amdgcn_wmma_f32_16x16x4_f32(
              /*neg_a=*/false, a, /*neg_b=*/false, b,
              /*c_mod=*/(short)0, acc[t], /*reuse_a=*/false, /*reuse_b=*/false);
        }
      }
    }
  }

  // ---- stage C tiles to LDS (wave-private region), then coalesced store ----
  float* so = &sO[wave * COUTP * 16];
  #pragma unroll
  for (int t = 0; t < NT; ++t) {
    #pragma unroll
    for (int r = 0; r < 8; ++r) {
      const int mr = (lane < 16) ? r : (r + 8); // ISA C/D: vgpr r -> M=r / M=r+8
      so[(t * 16 + m) * 16 + mr] = acc[t][r];
    }
  }
  // same-wave LDS ops are in-order; sO region is wave-private -> no barrier
  for (int idx = lane; idx < COUT * 16; idx += 32) {
    const int co = idx >> 4;
    const int mm = idx & 15;
    float v = so[idx] + bias[co];
    v = (ACT == 0) ? fmaxf(v, 0.f) : 1.f / (1.f + __expf(-v));
    y[(((size_t)n * COUT + co) * HH + h) * WW + w0 + mm] = v;
  }
}

// ---------------------------------------------------------------------------
// Weight reorder into GEMM-friendly [pos][ci][co] layout.
//   flip==0: src is Conv2d OIHW [O][I][3][3]
//   flip==1: src is ConvTranspose2d (I,O,3,3); effective conv weight is
//            IO-swapped + spatially flipped: src[i][o][2-kh][2-kw]
// ---------------------------------------------------------------------------
__global__ void wt_reorder(const float* __restrict__ src,
                           float* __restrict__ dst, int I, int O, int flip) {
  int idx = blockIdx.x * blockDim.x + threadIdx.x;
  int total = I * O * 9;
  if (idx >= total) return;
  int o   = idx % O;
  int t   = idx / O;
  int i   = t % I;
  int pos = t / I;
  int kh  = pos / 3;
  int kw  = pos - kh * 3;
  float v;
  if (flip) v = src[((i * O + o) * 3 + (2 - kh)) * 3 + (2 - kw)];
  else      v = src[((o * I + i) * 3 + kh) * 3 + kw];
  dst[idx] = v;
}

// ---------------------------------------------------------------------------
// Vector quantizer: rows are 8 *consecutive* floats of z (NCHW flatten, as in
// torch .view). Codebook (512x8, 16KB) + norms cached in LDS. Accumulates
// sum((q-z)^2) into lossAccum via one atomicAdd per block.
// ---------------------------------------------------------------------------
__global__ __launch_bounds__(256)
void vq_kernel(const float* __restrict__ z, const float* __restrict__ emb,
               float* __restrict__ q, float* __restrict__ lossAccum) {
  __shared__ float sE[512 * 8];
  __shared__ float sN[512];
  __shared__ float red[256];
  const int tid = threadIdx.x;

  for (int i = tid; i < 512 * 8; i += 256) sE[i] = emb[i];
  __syncthreads();
  for (int k = tid; k < 512; k += 256) {
    float s = 0.f;
    #pragma unroll
    for (int j = 0; j < 8; ++j) { float e = sE[k * 8 + j]; s += e * e; }
    sN[k] = s;
  }
  __syncthreads();

  const int row = blockIdx.x * 256 + tid;
  float f[8];
  float fn = 0.f;
  #pragma unroll
  for (int j = 0; j < 8; ++j) {
    f[j] = z[(size_t)row * 8 + j];
    fn += f[j] * f[j];
  }

  float bestd = 3.4e38f;
  int   besti = 0;
  for (int k = 0; k < 512; ++k) {
    float dot = 0.f;
    #pragma unroll
    for (int j = 0; j < 8; ++j) dot += f[j] * sE[k * 8 + j];
    float d = fn + sN[k] - 2.f * dot;
    if (d < bestd) { bestd = d; besti = k; }   // strict '<' keeps first argmin
  }

  float sq = 0.f;
  #pragma unroll
  for (int j = 0; j < 8; ++j) {
    float e = sE[besti * 8 + j];
    q[(size_t)row * 8 + j] = e;
    float dd = e - f[j];
    sq += dd * dd;
  }

  red[tid] = sq;
  __syncthreads();
  for (int s = 128; s > 0; s >>= 1) {
    if (tid < s) red[tid] += red[tid + s];
    __syncthreads();
  }
  if (tid == 0) atomicAdd(lossAccum, red[0]);
}

__global__ void finalize_loss(const float* __restrict__ lossAccum,
                              float* __restrict__ out) {
  // vq_loss = q_loss + 0.25*e_loss = 1.25 * mean((q - z)^2), numel = 16*8*128*128
  out[0] = 1.25f * lossAccum[0] / 2097152.0f;
}

// ---------------------------------------------------------------------------
extern "C" void kernel_launch(void* const* d_in, const int* in_sizes, int n_in,
                              void* d_out, int out_size, void* d_ws, size_t ws_size,
                              hipStream_t stream) {
  (void)in_sizes; (void)n_in; (void)out_size; (void)ws_size;

  const float* x   = (const float*)d_in[0];
  const float* e1w = (const float*)d_in[1];
  const float* e1b = (const float*)d_in[2];
  const float* e2w = (const float*)d_in[3];
  const float* e2b = (const float*)d_in[4];
  const float* e3w = (const float*)d_in[5];
  const float* e3b = (const float*)d_in[6];
  const float* emb = (const float*)d_in[7];
  const float* d1w = (const float*)d_in[8];
  const float* d1b = (const float*)d_in[9];
  const float* d2w = (const float*)d_in[10];
  const float* d2b = (const float*)d_in[11];
  const float* d3w = (const float*)d_in[12];
  const float* d3b = (const float*)d_in[13];

  char* ws = (char*)d_ws;
  size_t off = 0;
  auto alloc = [&](size_t bytes) -> void* {
    void* p = ws + off;
    off += (bytes + 255) & ~(size_t)255;
    return p;
  };
  float* bufA = (float*)alloc((size_t)16 * 128 * 128 * 128 * 4); // 134 MB
  float* bufB = (float*)alloc((size_t)16 *  64 * 128 * 128 * 4); //  67 MB
  float* z    = (float*)alloc((size_t)16 *   8 * 128 * 128 * 4); // 8.4 MB
  float* q    = (float*)alloc((size_t)16 *   8 * 128 * 128 * 4); // 8.4 MB
  float* w1r  = (float*)alloc((size_t)156 * 128 * 9 * 4);        // e1 [pos][ci][co]
  float* w2r  = (float*)alloc((size_t)128 *  64 * 9 * 4);
  float* w3r  = (float*)alloc((size_t) 64 *   8 * 9 * 4);
  float* w4r  = (float*)alloc((size_t)  8 *  64 * 9 * 4);        // d1
  float* w5r  = (float*)alloc((size_t) 64 * 128 * 9 * 4);
  float* w6r  = (float*)alloc((size_t)128 * 156 * 9 * 4);
  float* loss = (float*)alloc(256);

  hipMemsetAsync(loss, 0, sizeof(float), stream);

  // reorder all weights to [pos][ci][co] (decoder: fold flip + IO swap)
  wt_reorder<<<(156 * 128 * 9 + 255) / 256, 256, 0, stream>>>(e1w, w1r, 156, 128, 0);
  wt_reorder<<<(128 *  64 * 9 + 255) / 256, 256, 0, stream>>>(e2w, w2r, 128,  64, 0);
  wt_reorder<<<( 64 *   8 * 9 + 255) / 256, 256, 0, stream>>>(e3w, w3r,  64,   8, 0);
  wt_reorder<<<(  8 *  64 * 9 + 255) / 256, 256, 0, stream>>>(d1w, w4r,   8,  64, 1);
  wt_reorder<<<( 64 * 128 * 9 + 255) / 256, 256, 0, stream>>>(d2w, w5r,  64, 128, 1);
  wt_reorder<<<(128 * 156 * 9 + 255) / 256, 256, 0, stream>>>(d3w, w6r, 128, 156, 1);

  // 16 imgs * 128 rows * 8 strips = 16384 strips; 4 strips (waves) per block
  dim3 grid(4096), block(128);

  // encoder
  conv3x3_wmma<156, 128, 0><<<grid, block, 0, stream>>>(x,    w1r, e1b, bufA);
  conv3x3_wmma<128,  64, 0><<<grid, block, 0, stream>>>(bufA, w2r, e2b, bufB);
  conv3x3_wmma< 64,   8, 0><<<grid, block, 0, stream>>>(bufB, w3r, e3b, z);

  // quantizer: 262144 rows of 8
  vq_kernel<<<1024, 256, 0, stream>>>(z, emb, q, loss);

  // decoder
  conv3x3_wmma<  8,  64, 0><<<grid, block, 0, stream>>>(q,    w4r, d1b, bufB);
  conv3x3_wmma< 64, 128, 0><<<grid, block, 0, stream>>>(bufB, w5r, d2b, bufA);
  conv3x3_wmma<128, 156, 1><<<grid, block, 0, stream>>>(bufA, w6r, d3b, (float*)d_out);

  finalize_loss<<<1, 1, 0, stream>>>(loss, (float*)d_out + (size_t)16 * 156 * 128 * 128);
}